// AtomEmbedder_37434934952474
// MI455X (gfx1250) — compile-verified
//
#include <hip/hip_runtime.h>

#define F_IN 11
#define D 256

typedef float v2f __attribute__((ext_vector_type(2)));
typedef float v8f __attribute__((ext_vector_type(8)));

// ------------------------------------------------------------------ utility
__global__ void zero_int_kernel(int* __restrict__ p, int n) {
  int i = blockIdx.x * blockDim.x + threadIdx.x;
  if (i < n) p[i] = 0;
}

// count[i] = #edges with dst == i
__global__ void count_kernel(const int* __restrict__ dst, int* __restrict__ count, int e) {
  int i = blockIdx.x * blockDim.x + threadIdx.x;
  if (i < e) atomicAdd(&count[dst[i]], 1);
}

// Single-block exclusive scan: row_ptr[0..n] from count[0..n-1].
__global__ __launch_bounds__(1024) void scan_kernel(const int* __restrict__ count,
                                                    int* __restrict__ row_ptr, int n) {
  __shared__ int buf[1024];
  __shared__ int carry;
  if (threadIdx.x == 0) { carry = 0; row_ptr[0] = 0; }
  __syncthreads();
  for (int base = 0; base < n; base += 1024) {
    int i = base + (int)threadIdx.x;
    buf[threadIdx.x] = (i < n) ? count[i] : 0;
    __syncthreads();
#pragma unroll
    for (int off = 1; off < 1024; off <<= 1) {   // Hillis-Steele inclusive scan
      int t = (threadIdx.x >= (unsigned)off) ? buf[threadIdx.x - off] : 0;
      __syncthreads();
      buf[threadIdx.x] += t;
      __syncthreads();
    }
    int incl = buf[threadIdx.x] + carry;
    if (i < n) row_ptr[i + 1] = incl;
    __syncthreads();
    if (threadIdx.x == 1023) carry = incl;       // running total (tail zeros ok)
    __syncthreads();
  }
}

// dinv[i] = rsqrt(edge_deg + 1)   (the +1 is the self-loop)
__global__ void dinv_kernel(const int* __restrict__ row_ptr, float* __restrict__ dinv, int n) {
  int i = blockIdx.x * blockDim.x + threadIdx.x;
  if (i < n) dinv[i] = rsqrtf((float)(row_ptr[i + 1] - row_ptr[i]) + 1.0f);
}

// Scatter edge sources into destination-CSR order.
__global__ void fill_kernel(const int* __restrict__ src, const int* __restrict__ dst,
                            const int* __restrict__ row_ptr, int* __restrict__ cursor,
                            int* __restrict__ csr_src, int e) {
  int i = blockIdx.x * blockDim.x + threadIdx.x;
  if (i < e) {
    int d = dst[i];
    int pos = atomicAdd(&cursor[d], 1);
    csr_src[row_ptr[d] + pos] = src[i];
  }
}

// ------------------------------------------------------------- embed + relu
__global__ __launch_bounds__(D) void embed_relu_kernel(const float* __restrict__ x,
                                                       const float* __restrict__ W,
                                                       const float* __restrict__ b,
                                                       float* __restrict__ h) {
  __shared__ float xs[F_IN];
  const int node = blockIdx.x;
  const int c = threadIdx.x;
  if (c < F_IN) xs[c] = x[node * F_IN + c];
  __syncthreads();
  float acc = b[c];
#pragma unroll
  for (int k = 0; k < F_IN; ++k) acc = fmaf(xs[k], W[k * D + c], acc);
  h[(size_t)node * D + c] = fmaxf(acc, 0.0f);
}

// ----------------------------------------------------------- WMMA fp32 GEMM
// Out = A @ W ; A:[n,256], W:[256,256].  grid.x = n/16 row tiles, block = 128
// (4 waves); wave w computes the 16x64 strip at columns [64w, 64w+64) using
// V_WMMA_F32_16X16X4_F32 with 4 accumulators (A fragment reused 4x per step).
__global__ __launch_bounds__(128) void gemm_wmma_kernel(const float* __restrict__ A,
                                                        const float* __restrict__ W,
                                                        float* __restrict__ Out) {
  const int lane = threadIdx.x & 31;
  const int wave = threadIdx.x >> 5;
  const int row0 = blockIdx.x * 16;
  const int col0 = wave * 64;
  const int hi = lane >> 4;   // 0 -> K pair {k,k+1}, 1 -> K pair {k+2,k+3}
  const int l  = lane & 15;

  const float* arow = A + (size_t)(row0 + l) * D + 2 * hi;  // 8B aligned
  v8f acc0 = {}, acc1 = {}, acc2 = {}, acc3 = {};

#pragma unroll 4
  for (int k = 0; k < D; k += 4) {
    v2f a = *(const v2f*)(arow + k);                 // A[row0+l][k+2hi .. +1]
    const float* wk = W + (size_t)(k + 2 * hi) * D + col0 + l;
    v2f b0, b1, b2, b3;
    b0.x = wk[0];   b0.y = wk[D];                    // B: K rows k+2hi, k+1+2hi
    b1.x = wk[16];  b1.y = wk[D + 16];
    b2.x = wk[32];  b2.y = wk[D + 32];
    b3.x = wk[48];  b3.y = wk[D + 48];
    acc0 = __builtin_amdgcn_wmma_f32_16x16x4_f32(false, a, false, b0, (short)0, acc0, false, false);
    acc1 = __builtin_amdgcn_wmma_f32_16x16x4_f32(false, a, false, b1, (short)0, acc1, false, false);
    acc2 = __builtin_amdgcn_wmma_f32_16x16x4_f32(false, a, false, b2, (short)0, acc2, false, false);
    acc3 = __builtin_amdgcn_wmma_f32_16x16x4_f32(false, a, false, b3, (short)0, acc3, false, false);
  }

  float* orow = Out + (size_t)(row0 + 8 * hi) * D + col0 + l;
#pragma unroll
  for (int r = 0; r < 8; ++r) {
    orow[(size_t)r * D + 0]  = acc0[r];
    orow[(size_t)r * D + 16] = acc1[r];
    orow[(size_t)r * D + 32] = acc2[r];
    orow[(size_t)r * D + 48] = acc3[r];
  }
}

// --------------------------------------------------------- CSR gather (no atomics)
// out[i,:] = maybe_relu( hw[i,:]*dinv[i]^2 + b + sum_{s in N(i)} hw[s,:]*dinv[s]*dinv[i] )
__global__ __launch_bounds__(D) void gather_kernel(const float* __restrict__ hw,
                                                   const float* __restrict__ dinv,
                                                   const int* __restrict__ row_ptr,
                                                   const int* __restrict__ csr_src,
                                                   const float* __restrict__ b,
                                                   float* __restrict__ out,
                                                   int do_relu) {
  const int i = blockIdx.x;
  const int c = threadIdx.x;
  const float di = dinv[i];
  float acc = hw[(size_t)i * D + c] * (di * di) + b[c];  // self-loop + bias

  const int beg = row_ptr[i];
  const int end = row_ptr[i + 1];
  int j = beg;
  for (; j + 1 < end; j += 2) {            // uniform indices -> scalar loads
    const int s0 = csr_src[j];
    const int s1 = csr_src[j + 1];
    const float n0 = di * dinv[s0];
    const float n1 = di * dinv[s1];
    acc = fmaf(hw[(size_t)s0 * D + c], n0, acc);
    acc = fmaf(hw[(size_t)s1 * D + c], n1, acc);
  }
  if (j < end) {
    const int s0 = csr_src[j];
    acc = fmaf(hw[(size_t)s0 * D + c], di * dinv[s0], acc);
  }
  if (do_relu) acc = fmaxf(acc, 0.0f);
  out[(size_t)i * D + c] = acc;
}

// ---------------------------------------------------------------------------
extern "C" void kernel_launch(void* const* d_in, const int* in_sizes, int n_in,
                              void* d_out, int out_size, void* d_ws, size_t ws_size,
                              hipStream_t stream) {
  const float* x       = (const float*)d_in[0];
  const int*   edges   = (const int*)d_in[1];
  const float* W_embed = (const float*)d_in[2];
  const float* b_embed = (const float*)d_in[3];
  const float* W1      = (const float*)d_in[4];
  const float* b1      = (const float*)d_in[5];
  const float* W2      = (const float*)d_in[6];
  const float* b2      = (const float*)d_in[7];
  float* out = (float*)d_out;

  const int n = in_sizes[0] / F_IN;  // 50000 (multiple of 16)
  const int e = in_sizes[1] / 2;     // 300000
  const int* src = edges;            // edge_index[0]
  const int* dsn = edges + e;        // edge_index[1]

  // Workspace layout (256B-aligned slices):
  //   dinv [n] f32 | row_ptr [n+1] i32 | cursor [n] i32 | csr_src [e] i32
  //   h [n,D] f32  | hw [n,D] f32          (~104 MB total)
  char* ws = (char*)d_ws;
  size_t off = 0;
  auto align256 = [](size_t v) { return (v + 255) & ~(size_t)255; };
  float* dinv    = (float*)(ws + off); off = align256(off + (size_t)n * 4);
  int*   row_ptr = (int*)  (ws + off); off = align256(off + (size_t)(n + 1) * 4);
  int*   cursor  = (int*)  (ws + off); off = align256(off + (size_t)n * 4);
  int*   csr_src = (int*)  (ws + off); off = align256(off + (size_t)e * 4);
  float* h       = (float*)(ws + off); off = align256(off + (size_t)n * D * 4);
  float* hw      = (float*)(ws + off);

  const int nb = (n + 255) / 256;
  const int eb = (e + 255) / 256;

  // ---- build destination-CSR (reused by both conv layers) ----
  zero_int_kernel<<<nb, 256, 0, stream>>>(cursor, n);
  count_kernel<<<eb, 256, 0, stream>>>(dsn, cursor, e);
  scan_kernel<<<1, 1024, 0, stream>>>(cursor, row_ptr, n);
  dinv_kernel<<<nb, 256, 0, stream>>>(row_ptr, dinv, n);
  zero_int_kernel<<<nb, 256, 0, stream>>>(cursor, n);
  fill_kernel<<<eb, 256, 0, stream>>>(src, dsn, row_ptr, cursor, csr_src, e);

  // ---- embed + relu ----
  embed_relu_kernel<<<n, D, 0, stream>>>(x, W_embed, b_embed, h);

  // ---- GCN layer 1: hw = h@W1 ; gather -> h (relu fused) ----
  gemm_wmma_kernel<<<n / 16, 128, 0, stream>>>(h, W1, hw);
  gather_kernel<<<n, D, 0, stream>>>(hw, dinv, row_ptr, csr_src, b1, h, 1);

  // ---- GCN layer 2: hw = h@W2 ; gather -> d_out (no relu) ----
  gemm_wmma_kernel<<<n / 16, 128, 0, stream>>>(h, W2, hw);
  gather_kernel<<<n, D, 0, stream>>>(hw, dinv, row_ptr, csr_src, b2, out, 0);
}